// LJLKWholePoseScoringModule_60962765799899
// MI455X (gfx1250) — compile-verified
//
#include <hip/hip_runtime.h>
#include <hip/hip_bf16.h>

typedef __attribute__((ext_vector_type(2))) float v2f;
typedef __attribute__((ext_vector_type(8))) float v8f;

#define NPOSE 2
#define NBLK  64
#define NATB  32      // atoms per block
#define NATOM 2048    // NBLK * NATB
#define NT    128     // NATOM / 16 tiles
#define JT_PER_WAVE 4
#define WAVES_PER_BLOCK 8

// 1/(2*pi^1.5)
#define INV2PI32 0.08979344f
// c = 1/0.6 :  c^12 - 2c^6      and  12*(c^7 - c^13)
#define LJ_F0C   416.52683f
#define LJ_FP0C  (-8759.206f)

__global__ void zero_kernel(float* out, int n) {
    int i = blockIdx.x * blockDim.x + threadIdx.x;
    if (i < n) out[i] = 0.0f;
}

__global__ __launch_bounds__(256) void ljlk_wmma_kernel(
    const float* __restrict__ coords,            // (P,N,3)
    const int*   __restrict__ block_types,       // (P,B)
    const int*   __restrict__ min_block_bondsep, // (P,B,B)
    const int*   __restrict__ bt_n_atoms,        // (NBT,)
    const int*   __restrict__ bt_atom_types,     // (NBT,A)
    const int*   __restrict__ bt_path_distance,  // (NBT,A,A)
    const float* __restrict__ type_params,       // (NAT,9)
    const float* __restrict__ global_params,     // (3,)
    float*       __restrict__ out)               // (2,P) flat
{
    __shared__ float s_acc[2 * NPOSE];
    const int tid = threadIdx.x;
    if (tid < 2 * NPOSE) s_acc[tid] = 0.0f;
    __syncthreads();

    const int lane = tid & 31;
    const int lm   = lane & 15;   // column / row slot 0..15
    const int half = lane >> 4;   // 0: low half, 1: high half

    // wave id forced uniform (SGPR) so the whole task decomposition is scalar:
    // block_types/bt_n_atoms/min_block_bondsep become s_loads, and the
    // same-block branch is a plain s_cbranch with no EXEC juggling.
    const int wv  = __builtin_amdgcn_readfirstlane(tid >> 5);
    const int wid = blockIdx.x * WAVES_PER_BLOCK + wv;
    const int p   = wid / (NT * (NT / JT_PER_WAVE));
    const int rem = wid % (NT * (NT / JT_PER_WAVE));
    const int it  = rem / (NT / JT_PER_WAVE);
    const int jt0 = (rem % (NT / JT_PER_WAVE)) * JT_PER_WAVE;

    const float g_dis = global_params[0];
    const float g_oh  = global_params[1];
    const float g_h   = global_params[2];

    // ---------------- row-side (i) setup: independent of jt ----------------
    const int bi_blk = it >> 1;                         // tile -> block (uniform)
    const int bi_t   = block_types[p * NBLK + bi_blk];  // s_load
    const int nat_i  = bt_n_atoms[bi_t];                // s_load
    const int wi_ln  = ((it & 1) << 4) + lm;            // within-block idx of my row atom
    const int ai_ln  = it * 16 + lm;

    const float* cbase = coords + (size_t)p * NATOM * 3;
    const float cix = cbase[ai_ln * 3 + 0];
    const float ciy = cbase[ai_ln * 3 + 1];
    const float ciz = cbase[ai_ln * 3 + 2];
    const float ni  = cix * cix + ciy * ciy + ciz * ciz;

    const int ti_ln = bt_atom_types[bi_t * NATB + wi_ln];
    const float* pri = type_params + ti_ln * 9;
    const float r_il    = pri[0];
    const float wd_il   = pri[1];
    const float lam_il  = pri[3];
    const float ilam_il = __builtin_amdgcn_rcpf(lam_il);        // v_rcp_f32
    const float gi_il   = pri[2] * ilam_il * INV2PI32;
    const float vol_il  = pri[4];
    const int   fl_il   = (pri[5] > 0.f ? 1 : 0) | (pri[6] > 0.f ? 2 : 0) |
                          (pri[7] > 0.f ? 4 : 0) | (pri[8] > 0.f ? 8 : 0) |
                          ((wi_ln < nat_i) ? 16 : 0);

    // broadcast params of the 8 D-rows this lane owns (row m = g + 8*half)
    float row_r[8], row_wd[8], row_gi[8], row_ilam[8], row_vol[8];
    int   row_fl[8];
    v8f C;
    #pragma unroll
    for (int g = 0; g < 8; ++g) {
        const int src = g + 8 * half;
        row_r[g]    = __shfl(r_il,    src, 32);
        row_wd[g]   = __shfl(wd_il,   src, 32);
        row_gi[g]   = __shfl(gi_il,   src, 32);
        row_ilam[g] = __shfl(ilam_il, src, 32);
        row_vol[g]  = __shfl(vol_il,  src, 32);
        row_fl[g]   = __shfl(fl_il,   src, 32);
        C[g]        = __shfl(ni,      src, 32) + 1e-6f;   // |ci|^2 + eps
    }

    // A fragment (16x4 f32): lanes 0-15 -> (x,y), lanes 16-31 -> (z, 1)
    v2f Am;
    Am.x = half ? ciz  : cix;
    Am.y = half ? 1.0f : ciy;

    float acc_lj = 0.0f, acc_lk = 0.0f;

    for (int jj = 0; jj < JT_PER_WAVE; ++jj) {
        const int jt = jt0 + jj;
        if (jt < it) continue;   // uniform scalar branch, EXEC stays all-1s

        // ---------------- column-side (j) setup ----------------
        const int bj_blk = jt >> 1;                        // uniform
        const int bj_t   = block_types[p * NBLK + bj_blk]; // s_load
        const int nat_j  = bt_n_atoms[bj_t];               // s_load
        const int wj     = ((jt & 1) << 4) + lm;
        const int aj     = jt * 16 + lm;

        const float cjx = cbase[aj * 3 + 0];
        const float cjy = cbase[aj * 3 + 1];
        const float cjz = cbase[aj * 3 + 2];
        const float nj  = cjx * cjx + cjy * cjy + cjz * cjz;

        const int tj = bt_atom_types[bj_t * NATB + wj];
        const float* prj = type_params + tj * 9;
        const float r_j    = prj[0];
        const float wd_j   = prj[1];
        const float lam_j  = prj[3];
        const float ilam_j = __builtin_amdgcn_rcpf(lam_j);
        const float gi_j   = prj[2] * ilam_j * INV2PI32;
        const float vol_j  = prj[4];
        const int   fj     = (prj[5] > 0.f ? 1 : 0) | (prj[6] > 0.f ? 2 : 0) |
                             (prj[7] > 0.f ? 4 : 0) | (prj[8] > 0.f ? 8 : 0);
        const int   vj     = (wj < nat_j) ? 1 : 0;

        const bool same_blk = (bi_blk == bj_blk);          // uniform
        const int  inter_sep = min_block_bondsep[(p * NBLK + bi_blk) * NBLK + bj_blk]; // s_load
        const int* pd_base = bt_path_distance + bi_t * NATB * NATB;

        // B fragment (4x16): lanes 0-15 -> (-2x,-2y), lanes 16-31 -> (-2z,|cj|^2)
        v2f Bm;
        Bm.x = half ? -2.0f * cjz : -2.0f * cjx;
        Bm.y = half ? nj          : -2.0f * cjy;

        // D[m][n] = |ci|^2 + |cj|^2 - 2 ci.cj + 1e-6  (16x16 tile of d^2)
        v8f D = __builtin_amdgcn_wmma_f32_16x16x4_f32(
            false, Am, false, Bm, (short)0, C, false, false);

        // bond-separation weights: same-block path (rare, uniform branch) loads
        // the path-distance table; inter-block path is a single scalar value.
        int sep_g[8];
        if (same_blk) {
            #pragma unroll
            for (int g = 0; g < 8; ++g) {
                const int wi = ((it & 1) << 4) + g + 8 * half;
                sep_g[g] = pd_base[wi * NATB + wj];
            }
        } else {
            #pragma unroll
            for (int g = 0; g < 8; ++g) sep_g[g] = inter_sep;
        }

        #pragma unroll
        for (int g = 0; g < 8; ++g) {
            const int m  = g + 8 * half;
            const int ai = it * 16 + m;

            const int fi = row_fl[g];
            // combined pair mask: valid_i & valid_j & (ai < aj)
            const int ok = ((fi >> 4) & vj) & (ai < aj ? 1 : 0);
            const int sep = sep_g[g];
            float w = (sep >= 5) ? 1.0f : ((sep == 4) ? 0.2f : 0.0f);
            w = ok ? w : 0.0f;

            const float d2 = fmaxf(D[g], 1e-6f);   // guard cancellation on masked pairs
            const float d  = __builtin_amdgcn_sqrtf(d2);        // v_sqrt_f32

            // fade window
            float t = fminf(fmaxf((6.0f - d) * (1.0f / 1.5f), 0.0f), 1.0f);
            const float wf = w * (t * t * (3.0f - 2.0f * t));

            // ---- sigma selection (branchless, int bit masks) ----
            const float r_i = row_r[g];
            const int don_acc_i = ((fi & (fj >> 3)) | ((fi >> 3) & fj)) & 1;
            const int hyd_i     = ((fi | fj) >> 1) & 1;
            const int ph_i      = (((fi >> 2) & (fj >> 3)) | ((fi >> 3) & (fj >> 2))) & 1;
            const float sig_da = hyd_i ? g_oh : g_dis;
            float sigma = r_i + r_j;
            sigma = don_acc_i ? sig_da : sigma;
            sigma = ph_i ? g_h : sigma;
            const float isigma = __builtin_amdgcn_rcpf(sigma);

            // ---- LJ ----
            const float eps = __builtin_amdgcn_sqrtf(row_wd[g] * wd_j);
            const float d0  = 0.6f * sigma;
            const float q   = sigma * __builtin_amdgcn_rcpf(fmaxf(d, d0));
            const float q3  = q * q * q;
            const float lr6 = q3 * q3;
            const float lj_std = eps * (lr6 * lr6 - 2.0f * lr6);
            const float lj_lin = eps * (LJ_F0C + LJ_FP0C * isigma * (d - d0));
            const float lj  = (d < d0) ? lj_lin : lj_std;
            acc_lj += wf * lj;

            // ---- LK ----
            const float dlk  = fmaxf(d, 1.0f);
            const float idlk = __builtin_amdgcn_rcpf(dlk);
            const float xi   = (dlk - r_i) * row_ilam[g];
            const float xj   = (dlk - r_j) * ilam_j;
            const float lk   = (row_gi[g] * __expf(-xi * xi) * vol_j +
                                gi_j      * __expf(-xj * xj) * row_vol[g])
                               * (idlk * idlk);
            acc_lk += wf * lk;
        }
    }

    // wave32 butterfly reduction
    #pragma unroll
    for (int off = 16; off > 0; off >>= 1) {
        acc_lj += __shfl_xor(acc_lj, off, 32);
        acc_lk += __shfl_xor(acc_lk, off, 32);
    }
    if (lane == 0) {
        atomicAdd(&s_acc[p], acc_lj);
        atomicAdd(&s_acc[NPOSE + p], acc_lk);
    }
    __syncthreads();
    if (tid < 2 * NPOSE) atomicAdd(&out[tid], s_acc[tid]);
}

extern "C" void kernel_launch(void* const* d_in, const int* in_sizes, int n_in,
                              void* d_out, int out_size, void* d_ws, size_t ws_size,
                              hipStream_t stream) {
    (void)in_sizes; (void)n_in; (void)d_ws; (void)ws_size;
    const float* coords            = (const float*)d_in[0];
    const int*   block_types       = (const int*)d_in[1];
    const int*   min_block_bondsep = (const int*)d_in[2];
    const int*   bt_n_atoms        = (const int*)d_in[3];
    const int*   bt_atom_types     = (const int*)d_in[4];
    const int*   bt_path_distance  = (const int*)d_in[5];
    const float* type_params       = (const float*)d_in[6];
    const float* global_params     = (const float*)d_in[7];
    float* out = (float*)d_out;

    zero_kernel<<<1, 64, 0, stream>>>(out, out_size);

    const int total_waves = NPOSE * NT * (NT / JT_PER_WAVE);   // 8192
    const int blocks = total_waves / WAVES_PER_BLOCK;          // 1024
    ljlk_wmma_kernel<<<blocks, 256, 0, stream>>>(
        coords, block_types, min_block_bondsep, bt_n_atoms,
        bt_atom_types, bt_path_distance, type_params, global_params, out);
}